// Gemma3Main_15573551415429
// MI455X (gfx1250) — compile-verified
//
#include <hip/hip_runtime.h>
#include <math.h>

// ---- model constants ----
#define LNUM 26
#define TT   16
#define DD   768
#define NHH  8
#define NGG  2
#define GQ   4          // NH/NG
#define HD   128
#define FFD  3072
#define VV   32000
#define SCL  4096
#define SEQ  (SCL + TT) // 4112
#define EPS  1e-6f
#define PSTRIDE 4116    // padded LDS row stride for probs (bank-conflict free)

typedef float v2f __attribute__((ext_vector_type(2)));
typedef float v8f __attribute__((ext_vector_type(8)));

__device__ __forceinline__ float warp_sum(float v) {
  for (int o = 16; o > 0; o >>= 1) v += __shfl_xor(v, o, 32);
  return v;
}
__device__ __forceinline__ float warp_max(float v) {
  for (int o = 16; o > 0; o >>= 1) v = fmaxf(v, __shfl_xor(v, o, 32));
  return v;
}

// ---------------------------------------------------------------------------
// RMSNorm over 16 rows (one wave per row, blockDim = 512)
__global__ void rmsnorm_rows(const float* __restrict__ x, const float* __restrict__ w,
                             float* __restrict__ out, int cols) {
  const int row  = threadIdx.x >> 5;
  const int lane = threadIdx.x & 31;
  const float* xr = x + (size_t)row * cols;
  float ss = 0.f;
  for (int c = lane; c < cols; c += 32) { float v = xr[c]; ss += v * v; }
  ss = warp_sum(ss);
  const float rs = rsqrtf(ss / (float)cols + EPS);
  float* orow = out + (size_t)row * cols;
  for (int c = lane; c < cols; c += 32) orow[c] = xr[c] * rs * (1.f + w[c]);
}

// h += rms_norm(y, w)
__global__ void residual_add_norm(float* __restrict__ h, const float* __restrict__ y,
                                  const float* __restrict__ w, int cols) {
  const int row  = threadIdx.x >> 5;
  const int lane = threadIdx.x & 31;
  const float* yr = y + (size_t)row * cols;
  float ss = 0.f;
  for (int c = lane; c < cols; c += 32) { float v = yr[c]; ss += v * v; }
  ss = warp_sum(ss);
  const float rs = rsqrtf(ss / (float)cols + EPS);
  float* hr = h + (size_t)row * cols;
  for (int c = lane; c < cols; c += 32) hr[c] += yr[c] * rs * (1.f + w[c]);
}

// ---------------------------------------------------------------------------
// Generic skinny GEMM: C[16 x N] = X[16 x K] * W[N x K]^T
// One wave per 16-column tile, V_WMMA_F32_16X16X4_F32 along K.
// Lane half hi owns K-slots {2hi, 2hi+1} for BOTH operands -> every operand is
// one contiguous global_load_b64, no selects in the streaming loop.
__global__ void __launch_bounds__(32)
gemm16_wmma(const float* __restrict__ X, int ldx,
            const float* __restrict__ W, int ldw,
            float* __restrict__ C, int ldc, int K) {
  const int lane = threadIdx.x & 31;
  const int m  = lane & 15;   // A row / B column (within tile)
  const int hi = lane >> 4;   // lane half -> K interleave slot
  const int n0 = blockIdx.x << 4;
  const float* xrow = X + (size_t)m * ldx + 2 * hi;        // A(M=m, K=k0+2hi..+1)
  const float* wrow = W + (size_t)(n0 + m) * ldw + 2 * hi; // B(K=k0+2hi..+1, N=n0+m)
  v8f acc = {};
  for (int k0 = 0; k0 < K; k0 += 4) {
    v2f a = *(const v2f*)(xrow + k0);
    v2f b = *(const v2f*)(wrow + k0);
    acc = __builtin_amdgcn_wmma_f32_16x16x4_f32(false, a, false, b,
                                                (short)0, acc, false, false);
  }
  for (int j = 0; j < 8; ++j)
    C[(size_t)(j + 8 * hi) * ldc + n0 + m] = acc[j];
}

// ---------------------------------------------------------------------------
// Split QKV, apply q/k RMSNorm + RoPE, scatter K/V new-slices (into d_out).
// grid = 12 blocks (NG * (G+2) slots), blockDim = 512 (wave per token).
__global__ void qkv_postprocess(const float* __restrict__ qkv,   // (T, NG*6*HD)
                                const float* __restrict__ qnw,
                                const float* __restrict__ knw,
                                const float* __restrict__ cosv,  // (T, HD)
                                const float* __restrict__ sinv,
                                float* __restrict__ qout,        // (NH, T, HD)
                                float* __restrict__ kslice,      // (NG, T, HD)
                                float* __restrict__ vslice) {    // (NG, HD, T)
  const int unit = blockIdx.x;
  const int g    = unit / 6;
  const int slot = unit % 6;
  const int t    = threadIdx.x >> 5;
  const int lane = threadIdx.x & 31;
  const float* src = qkv + (size_t)t * (NGG * 6 * HD) + (size_t)(g * 6 + slot) * HD;
  float x[4];
  float ss = 0.f;
  for (int i = 0; i < 4; ++i) { x[i] = src[lane + 32 * i]; ss += x[i] * x[i]; }
  if (slot == 5) {             // V: transpose copy (g, d, t)
    for (int i = 0; i < 4; ++i) {
      int d = lane + 32 * i;
      vslice[((size_t)g * HD + d) * TT + t] = x[i];
    }
    return;
  }
  ss = warp_sum(ss);
  const float rs = rsqrtf(ss / (float)HD + EPS);
  const float* nw = (slot == 4) ? knw : qnw;
  float xn[4];
  for (int i = 0; i < 4; ++i) {
    int d = lane + 32 * i;
    xn[i] = x[i] * rs * (1.f + nw[d]);
  }
  for (int i = 0; i < 4; ++i) {
    int d = lane + 32 * i;
    const float c = cosv[t * HD + d], s = sinv[t * HD + d];
    const float rot = (d < 64) ? -xn[i ^ 2] : xn[i ^ 2];  // partner at d +/- 64
    const float o = xn[i] * c + rot * s;
    if (slot == 4) kslice[((size_t)g * TT + t) * HD + d] = o;
    else           qout[(((size_t)(g * GQ + slot)) * TT + t) * HD + d] = o;
  }
}

// ---------------------------------------------------------------------------
// gelu(gate) * up
__global__ void gelu_mul(const float* __restrict__ g, const float* __restrict__ u,
                         float* __restrict__ o, int n) {
  int i = blockIdx.x * blockDim.x + threadIdx.x;
  if (i < n) {
    float x = g[i];
    float th = tanhf(0.7978845608028654f * (x + 0.044715f * x * x * x));
    o[i] = 0.5f * x * (1.f + th) * u[i];
  }
}

// ---------------------------------------------------------------------------
// One workgroup (8 waves) per head. Full 16x4112 score matrix lives in LDS
// (16 * 4116 * 4B = 263 KB of CDNA5's 320 KB WGP LDS).
//   pass 1: scores = softcap(q.K^T * scale) + mask  -> LDS   (WMMA, K stream)
//   pass 2: row softmax in LDS
//   pass 3: out = probs * V^T                        (WMMA, A from LDS via ds)
__global__ void __launch_bounds__(256)
attention_head(const float* __restrict__ q,     // (NH, T, HD)
               const float* __restrict__ kc,    // (NG, SC, HD)
               const float* __restrict__ vc,    // (NG, HD, SC)
               const float* __restrict__ knew,  // (NG, T, HD)
               const float* __restrict__ vnew,  // (NG, HD, T)
               const float* __restrict__ mask,  // (T, SEQ)
               float* __restrict__ out) {       // (T, NH*HD)
  extern __shared__ float probs[];              // 16 * PSTRIDE
  const int head = blockIdx.x;
  const int g    = head / GQ;
  const int wave = threadIdx.x >> 5;
  const int lane = threadIdx.x & 31;
  const int m    = lane & 15;
  const int hi   = lane >> 4;
  const float scale = 0.08838834764831845f;     // 1/sqrt(128)
  const float inv_cap = 1.f / 50.f;

  // ---- pass 1: score tiles (16 x 16), distributed over waves ----
  const float* qrow = q + ((size_t)head * TT + m) * HD + 2 * hi;
  const int ntiles = SEQ / 16;                  // 257
  for (int st = wave; st < ntiles; st += 8) {
    const int s0 = st * 16;
    const int s  = s0 + m;
    const float* krow = 2 * hi + ((s < SCL)
        ? (kc   + ((size_t)g * SCL + s) * HD)
        : (knew + ((size_t)g * TT + (s - SCL)) * HD));
    v8f acc = {};
    for (int k0 = 0; k0 < HD; k0 += 4) {
      v2f a = *(const v2f*)(qrow + k0);
      v2f b = *(const v2f*)(krow + k0);
      acc = __builtin_amdgcn_wmma_f32_16x16x4_f32(false, a, false, b,
                                                  (short)0, acc, false, false);
    }
    for (int j = 0; j < 8; ++j) {
      const int row = j + 8 * hi;               // query t
      const int col = s0 + m;
      const float sc = tanhf(acc[j] * scale * inv_cap) * 50.f + mask[row * SEQ + col];
      probs[row * PSTRIDE + col] = sc;
    }
  }
  __syncthreads();

  // ---- pass 2: softmax (wave w -> rows 2w, 2w+1) ----
  for (int r = wave * 2; r < wave * 2 + 2; ++r) {
    float* prow = probs + (size_t)r * PSTRIDE;
    float mx = -1e30f;
    for (int i = lane; i < SEQ; i += 32) mx = fmaxf(mx, prow[i]);
    mx = warp_max(mx);
    float sum = 0.f;
    for (int i = lane; i < SEQ; i += 32) sum += __expf(prow[i] - mx);
    sum = warp_sum(sum);
    const float inv = 1.f / sum;
    for (int i = lane; i < SEQ; i += 32) prow[i] = __expf(prow[i] - mx) * inv;
  }
  __syncthreads();

  // ---- pass 3: attn = probs x V^T, wave w -> d-tile [16w, 16w+16) ----
  const int d0 = wave * 16;
  const int d  = d0 + m;
  const float* vrow_c = vc   + ((size_t)g * HD + d) * SCL + 2 * hi;
  const float* vrow_n = vnew + ((size_t)g * HD + d) * TT  + 2 * hi;
  const float* prow = probs + (size_t)m * PSTRIDE + 2 * hi;   // A row = query m
  v8f acc = {};
  for (int s0 = 0; s0 < SCL; s0 += 4) {          // KV-cache part (branch-free)
    v2f a = *(const v2f*)(prow + s0);
    v2f b = *(const v2f*)(vrow_c + s0);
    acc = __builtin_amdgcn_wmma_f32_16x16x4_f32(false, a, false, b,
                                                (short)0, acc, false, false);
  }
  for (int s0 = SCL; s0 < SEQ; s0 += 4) {        // new-token part
    v2f a = *(const v2f*)(prow + s0);
    v2f b = *(const v2f*)(vrow_n + (s0 - SCL));
    acc = __builtin_amdgcn_wmma_f32_16x16x4_f32(false, a, false, b,
                                                (short)0, acc, false, false);
  }
  for (int j = 0; j < 8; ++j) {
    const int row = j + 8 * hi;                 // query t
    out[(size_t)row * (NHH * HD) + head * HD + d0 + m] = acc[j];
  }
}

// ---------------------------------------------------------------------------
extern "C" void kernel_launch(void* const* d_in, const int* in_sizes, int n_in,
                              void* d_out, int out_size, void* d_ws, size_t ws_size,
                              hipStream_t stream) {
  (void)in_sizes; (void)n_in; (void)out_size; (void)ws_size;
  const float* emb       = (const float*)d_in[0];
  const float* mask_g    = (const float*)d_in[1];
  const float* mask_l    = (const float*)d_in[2];
  const float* pe_cos    = (const float*)d_in[3];
  const float* pe_sin    = (const float*)d_in[4];
  const float* pel_cos   = (const float*)d_in[5];
  const float* pel_sin   = (const float*)d_in[6];
  const float* kcache    = (const float*)d_in[7];
  const float* vcache    = (const float*)d_in[8];
  const float* pan       = (const float*)d_in[9];
  const float* qkvw      = (const float*)d_in[10];
  const float* qnw       = (const float*)d_in[11];
  const float* knw       = (const float*)d_in[12];
  const float* ow        = (const float*)d_in[13];
  const float* postw     = (const float*)d_in[14];
  const float* pfn       = (const float*)d_in[15];
  const float* gw        = (const float*)d_in[16];
  const float* uw        = (const float*)d_in[17];
  const float* dw        = (const float*)d_in[18];
  const float* pofn      = (const float*)d_in[19];
  const float* fnw       = (const float*)d_in[20];
  const float* lmw       = (const float*)d_in[21];

  // workspace layout (floats)
  float* ws    = (float*)d_ws;
  float* h     = ws;                 // 16*768
  float* xn    = h     + TT * DD;    // 16*768
  float* qkv   = xn    + TT * DD;    // 16*1536
  float* qbuf  = qkv   + TT * 1536;  // 8*16*128
  float* attno = qbuf  + NHH * TT * HD;  // 16*1024
  float* y     = attno + TT * NHH * HD;  // 16*768
  float* gate  = y     + TT * DD;    // 16*3072
  float* up    = gate  + TT * FFD;   // 16*3072
  float* ffact = up    + TT * FFD;   // 16*3072
  float* ffd   = ffact + TT * FFD;   // 16*768

  // output layout: logits | k_slices (L,NG,T,HD) | v_slices (L,NG,HD,T)
  float* logits  = (float*)d_out;
  float* kslices = logits  + (size_t)TT * VV;
  float* vslices = kslices + (size_t)LNUM * NGG * TT * HD;

  // allow >64KB dynamic LDS for the attention kernel
  hipFuncSetAttribute((const void*)attention_head,
                      hipFuncAttributeMaxDynamicSharedMemorySize,
                      16 * PSTRIDE * (int)sizeof(float));

  // residual init
  hipMemcpyAsync(h, emb, (size_t)TT * DD * sizeof(float),
                 hipMemcpyDeviceToDevice, stream);

  for (int l = 0; l < LNUM; ++l) {
    const bool local = ((l + 1) % 6) == 0;
    const float* cosv = local ? pel_cos : pe_cos;
    const float* sinv = local ? pel_sin : pe_sin;
    const float* mask = local ? mask_l : mask_g;
    float* kslice = kslices + (size_t)l * NGG * TT * HD;
    float* vslice = vslices + (size_t)l * NGG * HD * TT;

    rmsnorm_rows<<<1, 512, 0, stream>>>(h, pan + (size_t)l * DD, xn, DD);

    gemm16_wmma<<<(NGG * 6 * HD) / 16, 32, 0, stream>>>(
        xn, DD, qkvw + (size_t)l * 1536 * DD, DD, qkv, 1536, DD);

    qkv_postprocess<<<12, 512, 0, stream>>>(
        qkv, qnw + (size_t)l * HD, knw + (size_t)l * HD, cosv, sinv,
        qbuf, kslice, vslice);

    attention_head<<<NHH, 256, 16 * PSTRIDE * sizeof(float), stream>>>(
        qbuf,
        kcache + (size_t)l * NGG * SCL * HD,
        vcache + (size_t)l * NGG * HD * SCL,
        kslice, vslice, mask, attno);

    gemm16_wmma<<<DD / 16, 32, 0, stream>>>(
        attno, NHH * HD, ow + (size_t)l * DD * NHH * HD, NHH * HD, y, DD, NHH * HD);

    residual_add_norm<<<1, 512, 0, stream>>>(h, y, postw + (size_t)l * DD, DD);

    rmsnorm_rows<<<1, 512, 0, stream>>>(h, pfn + (size_t)l * DD, xn, DD);

    gemm16_wmma<<<FFD / 16, 32, 0, stream>>>(
        xn, DD, gw + (size_t)l * FFD * DD, DD, gate, FFD, DD);
    gemm16_wmma<<<FFD / 16, 32, 0, stream>>>(
        xn, DD, uw + (size_t)l * FFD * DD, DD, up, FFD, DD);

    gelu_mul<<<(TT * FFD + 255) / 256, 256, 0, stream>>>(gate, up, ffact, TT * FFD);

    gemm16_wmma<<<DD / 16, 32, 0, stream>>>(
        ffact, FFD, dw + (size_t)l * DD * FFD, FFD, ffd, DD, FFD);

    residual_add_norm<<<1, 512, 0, stream>>>(h, ffd, pofn + (size_t)l * DD, DD);
  }

  rmsnorm_rows<<<1, 512, 0, stream>>>(h, fnw, xn, DD);
  gemm16_wmma<<<VV / 16, 32, 0, stream>>>(xn, DD, lmw, DD, logits, VV, DD);
}